// TargetGraphEncoder_27427661152648
// MI455X (gfx1250) — compile-verified
//
#include <hip/hip_runtime.h>
#include <hip/hip_bf16.h>

typedef float v2f __attribute__((ext_vector_type(2)));
typedef float v8f __attribute__((ext_vector_type(8)));

#define GCN_H 128
#define BN_EPS 1e-5f

// ---------------------------------------------------------------------------
// Degree / dinv kernels
// ---------------------------------------------------------------------------
__global__ void init_deg_kernel(float* __restrict__ deg, int N) {
    int i = blockIdx.x * blockDim.x + threadIdx.x;
    if (i < N) deg[i] = 1.0f;  // self loop contributes 1 to every node
}

__global__ void deg_edges_kernel(const int* __restrict__ col, float* __restrict__ deg, int E) {
    int e = blockIdx.x * blockDim.x + threadIdx.x;
    if (e < E) unsafeAtomicAdd(&deg[col[e]], 1.0f);
}

__global__ void dinv_kernel(float* __restrict__ d, int N) {
    int i = blockIdx.x * blockDim.x + threadIdx.x;
    if (i < N) d[i] = __frsqrt_rn(d[i]);  // deg >= 1 always (self loops)
}

// ---------------------------------------------------------------------------
// GEMM: Out[N,128] = A[N,128] @ W[128,128]  via V_WMMA_F32_16X16X4_F32
// Block = 256 threads = 8 waves; wave w computes rows [16*blockIdx.x, +16),
// cols [16*w, +16). K=128 -> 32 WMMA steps of K=4, fully unrolled (no loop
// SALU/branch overhead; long load clauses). No divergence -> EXEC all 1s.
// ---------------------------------------------------------------------------
__global__ __launch_bounds__(256) void gemm_wmma_f32(const float* __restrict__ A,
                                                     const float* __restrict__ W,
                                                     float* __restrict__ Out) {
    const int wave  = threadIdx.x >> 5;
    const int lane  = threadIdx.x & 31;
    const int lhalf = lane >> 4;   // 0: lanes 0-15, 1: lanes 16-31
    const int lm    = lane & 15;
    const int m0 = blockIdx.x * 16;
    const int n0 = wave * 16;

    v8f c = {};
    const float* arow = A + (size_t)(m0 + lm) * GCN_H + 2 * lhalf;  // lane's K pair base
    const float* bcol = W + (size_t)(2 * lhalf) * GCN_H + n0 + lm;  // lane's B column base
    #pragma unroll
    for (int k0 = 0; k0 < GCN_H; k0 += 4) {
        // A 16x4 fragment: lane holds A[M=lm][ka], A[M=lm][ka+1]  (even addr -> b64 load)
        v2f a = *(const v2f*)(arow + k0);
        // B 4x16 fragment: lane holds W[ka][n0+lm], W[ka+1][n0+lm]
        v2f b;
        b.x = bcol[(size_t)k0 * GCN_H];
        b.y = bcol[(size_t)(k0 + 1) * GCN_H];
        c = __builtin_amdgcn_wmma_f32_16x16x4_f32(false, a, false, b, (short)0, c, false, false);
    }
    // D layout: VGPR r -> row (m0 + r + 8*lhalf), col (n0 + lm)
    float* orow = Out + (size_t)(m0 + 8 * lhalf) * GCN_H + n0 + lm;
    #pragma unroll
    for (int r = 0; r < 8; ++r) orow[(size_t)r * GCN_H] = c[r];
}

// ---------------------------------------------------------------------------
// Edge scatter: out[col[e]][:] += tmp[row[e]][:] * dinv[row]*dinv[col]
// One wave32 per edge; lane handles float4 chunk -> 512B coalesced gather,
// 128 hardware global_atomic_add_f32 per edge (agg buffer fits in 192MB L2).
// ---------------------------------------------------------------------------
__global__ __launch_bounds__(256) void edge_scatter_kernel(const float* __restrict__ tmp,
                                                           const int* __restrict__ row,
                                                           const int* __restrict__ col,
                                                           const float* __restrict__ dinv,
                                                           float* __restrict__ out, int E) {
    const int e = blockIdx.x * (blockDim.x >> 5) + (threadIdx.x >> 5);
    if (e >= E) return;
    const int lane = threadIdx.x & 31;
    const int r = row[e];
    const int c = col[e];
    const float nrm = dinv[r] * dinv[c];
    const float4 v = ((const float4*)(tmp + (size_t)r * GCN_H))[lane];
    float* dst = out + (size_t)c * GCN_H + lane * 4;
    unsafeAtomicAdd(dst + 0, v.x * nrm);
    unsafeAtomicAdd(dst + 1, v.y * nrm);
    unsafeAtomicAdd(dst + 2, v.z * nrm);
    unsafeAtomicAdd(dst + 3, v.w * nrm);
}

// ---------------------------------------------------------------------------
// Epilogue (layers 1,2): agg += self-loop + bias, then BN(eval) + ReLU, in place.
// ---------------------------------------------------------------------------
__global__ void bn_relu_epilogue_kernel(float* __restrict__ agg,
                                        const float* __restrict__ tmp,
                                        const float* __restrict__ dinv,
                                        const float* __restrict__ bias,
                                        const float* __restrict__ g,
                                        const float* __restrict__ be,
                                        const float* __restrict__ rm,
                                        const float* __restrict__ rv, int N) {
    size_t idx = (size_t)blockIdx.x * blockDim.x + threadIdx.x;
    if (idx >= (size_t)N * GCN_H) return;
    const int f = (int)(idx & (GCN_H - 1));
    const size_t i = idx >> 7;
    const float di = dinv[i];
    float val = agg[idx] + tmp[idx] * di * di + bias[f];
    val = (val - rm[f]) * __frsqrt_rn(rv[f] + BN_EPS) * g[f] + be[f];
    agg[idx] = fmaxf(val, 0.0f);
}

// ---------------------------------------------------------------------------
// Final layer epilogue + global_add_pool into d_out[G,128]
// ---------------------------------------------------------------------------
__global__ void pool_kernel(const float* __restrict__ agg,
                            const float* __restrict__ tmp,
                            const float* __restrict__ dinv,
                            const float* __restrict__ bias,
                            const int* __restrict__ batch,
                            float* __restrict__ out, int N) {
    size_t idx = (size_t)blockIdx.x * blockDim.x + threadIdx.x;
    if (idx >= (size_t)N * GCN_H) return;
    const int f = (int)(idx & (GCN_H - 1));
    const size_t i = idx >> 7;
    const float di = dinv[i];
    const float val = agg[idx] + tmp[idx] * di * di + bias[f];
    unsafeAtomicAdd(&out[(size_t)batch[i] * GCN_H + f], val);
}

// ---------------------------------------------------------------------------
// Host launcher
// ---------------------------------------------------------------------------
extern "C" void kernel_launch(void* const* d_in, const int* in_sizes, int n_in,
                              void* d_out, int out_size, void* d_ws, size_t ws_size,
                              hipStream_t stream) {
    const float* x    = (const float*)d_in[0];
    const int*   ei   = (const int*)d_in[1];
    const int*   batch= (const int*)d_in[2];
    const float* W1   = (const float*)d_in[3];
    const float* b1   = (const float*)d_in[4];
    const float* W2   = (const float*)d_in[5];
    const float* b2   = (const float*)d_in[6];
    const float* W3   = (const float*)d_in[7];
    const float* b3   = (const float*)d_in[8];
    const float* g1   = (const float*)d_in[9];
    const float* be1  = (const float*)d_in[10];
    const float* rm1  = (const float*)d_in[11];
    const float* rv1  = (const float*)d_in[12];
    const float* g2   = (const float*)d_in[13];
    const float* be2  = (const float*)d_in[14];
    const float* rm2  = (const float*)d_in[15];
    const float* rv2  = (const float*)d_in[16];

    const int N = in_sizes[0] / GCN_H;
    const int E = in_sizes[1] / 2;
    const int* row = ei;
    const int* col = ei + E;

    // Workspace layout: dinv [N] | bufA [N*128] | bufB [N*128]
    char* ws = (char*)d_ws;
    size_t off = 0;
    float* dinv = (float*)(ws + off); off += ((size_t)N * sizeof(float) + 255) & ~(size_t)255;
    float* bufA = (float*)(ws + off); off += (size_t)N * GCN_H * sizeof(float);
    float* bufB = (float*)(ws + off); off += (size_t)N * GCN_H * sizeof(float);
    (void)ws_size; (void)n_in;

    const size_t hBytes = (size_t)N * GCN_H * sizeof(float);
    const int nfThreads = 256;
    const int nfBlocksNH = (int)(((size_t)N * GCN_H + nfThreads - 1) / nfThreads);
    const int gemmBlocks = N / 16;                       // N % 16 == 0
    const int edgeBlocks = (E + 7) / 8;                  // 8 edges (waves) per block

    // Degrees -> dinv
    hipLaunchKernelGGL(init_deg_kernel, dim3((N + 255) / 256), dim3(256), 0, stream, dinv, N);
    hipLaunchKernelGGL(deg_edges_kernel, dim3((E + 255) / 256), dim3(256), 0, stream, col, dinv, E);
    hipLaunchKernelGGL(dinv_kernel, dim3((N + 255) / 256), dim3(256), 0, stream, dinv, N);

    hipMemsetAsync(d_out, 0, (size_t)out_size * sizeof(float), stream);

    // ---- Layer 1 ----
    hipLaunchKernelGGL(gemm_wmma_f32, dim3(gemmBlocks), dim3(256), 0, stream, x, W1, bufA);
    hipMemsetAsync(bufB, 0, hBytes, stream);
    hipLaunchKernelGGL(edge_scatter_kernel, dim3(edgeBlocks), dim3(256), 0, stream,
                       bufA, row, col, dinv, bufB, E);
    hipLaunchKernelGGL(bn_relu_epilogue_kernel, dim3(nfBlocksNH), dim3(nfThreads), 0, stream,
                       bufB, bufA, dinv, b1, g1, be1, rm1, rv1, N);

    // ---- Layer 2 ---- (h1 in bufB becomes dead after the GEMM reads it)
    hipLaunchKernelGGL(gemm_wmma_f32, dim3(gemmBlocks), dim3(256), 0, stream, bufB, W2, bufA);
    hipMemsetAsync(bufB, 0, hBytes, stream);
    hipLaunchKernelGGL(edge_scatter_kernel, dim3(edgeBlocks), dim3(256), 0, stream,
                       bufA, row, col, dinv, bufB, E);
    hipLaunchKernelGGL(bn_relu_epilogue_kernel, dim3(nfBlocksNH), dim3(nfThreads), 0, stream,
                       bufB, bufA, dinv, b2, g2, be2, rm2, rv2, N);

    // ---- Layer 3 + pooling ----
    hipLaunchKernelGGL(gemm_wmma_f32, dim3(gemmBlocks), dim3(256), 0, stream, bufB, W3, bufA);
    hipMemsetAsync(bufB, 0, hBytes, stream);
    hipLaunchKernelGGL(edge_scatter_kernel, dim3(edgeBlocks), dim3(256), 0, stream,
                       bufA, row, col, dinv, bufB, E);
    hipLaunchKernelGGL(pool_kernel, dim3(nfBlocksNH), dim3(nfThreads), 0, stream,
                       bufB, bufA, dinv, b3, batch, (float*)d_out, N);
}